// dgl_gcn_1099511628218
// MI455X (gfx1250) — compile-verified
//
#include <hip/hip_runtime.h>

typedef float v2f __attribute__((ext_vector_type(2)));
typedef float v8f __attribute__((ext_vector_type(8)));

#define D_IN  128
#define D_HID 128
#define D_OUT 64

// ---------------------------------------------------------------------------
// Utility kernels
// ---------------------------------------------------------------------------
__global__ void zero_f32(float* __restrict__ p, int n) {
    int i = blockIdx.x * blockDim.x + threadIdx.x;
    if (i < n) p[i] = 0.0f;
}

// Count degrees of both graphs in one pass (4 float atomics per edge).
__global__ void degree_kernel(const int* __restrict__ src1, const int* __restrict__ dst1,
                              const int* __restrict__ src2, const int* __restrict__ dst2,
                              float* __restrict__ rout1, float* __restrict__ rin1,
                              float* __restrict__ rout2, float* __restrict__ rin2, int E) {
    int i = blockIdx.x * blockDim.x + threadIdx.x;
    if (i >= E) return;
    atomicAdd(&rout1[src1[i]], 1.0f);
    atomicAdd(&rin1 [dst1[i]], 1.0f);
    atomicAdd(&rout2[src2[i]], 1.0f);
    atomicAdd(&rin2 [dst2[i]], 1.0f);
}

// deg -> rsqrt(max(deg, 1))
__global__ void rsqrt_clamp(float* __restrict__ p, int n) {
    int i = blockIdx.x * blockDim.x + threadIdx.x;
    if (i < n) p[i] = rsqrtf(fmaxf(p[i], 1.0f));
}

// out[i*D + f] = (attn0 + attn1) * bias[f]   (accumulator init; D is pow2)
__global__ void bias_fill(float* __restrict__ out, const float* __restrict__ bias,
                          const float* __restrict__ attn, int total, int dmask) {
    int i = blockIdx.x * blockDim.x + threadIdx.x;
    if (i >= total) return;
    out[i] = (attn[0] + attn[1]) * bias[i & dmask];
}

// ---------------------------------------------------------------------------
// fp32 WMMA GEMM: C[N,D] = A[N,K] @ B[K,D].  One wave (32 lanes) per 16-row
// strip; NT = D/16 column tiles, A fragment reused across all tiles.
// V_WMMA_F32_16X16X4_F32 layouts (ISA 7.12.2):
//   A (16x4):  lane l -> rows m0+(l&15), K = k + 2*(l>=16) + {0,1}
//   B (4x16):  lane l -> col  n0+(l&15), K = k + 2*(l>=16) + {0,1}
//   C/D:       vgpr v, lane l -> row m0 + v + 8*(l>=16), col n0+(l&15)
// ---------------------------------------------------------------------------
template <int NT>
__global__ __launch_bounds__(32)
void gemm_wmma_f32(const float* __restrict__ A, const float* __restrict__ B,
                   float* __restrict__ C, int K, int D) {
    const int m0   = blockIdx.x * 16;
    const int lane = threadIdx.x & 31;
    const int half = lane >> 4;      // 0 or 1
    const int lm   = lane & 15;

    v8f acc[NT];
#pragma unroll
    for (int t = 0; t < NT; ++t) acc[t] = (v8f){};

    const float* arow = A + (size_t)(m0 + lm) * K + 2 * half;

    for (int k = 0; k < K; k += 4) {
        v2f a;
        a.x = arow[k + 0];
        a.y = arow[k + 1];
#pragma unroll
        for (int t = 0; t < NT; ++t) {
            const float* bp = B + (size_t)(k + 2 * half) * D + t * 16 + lm;
            v2f b;
            b.x = bp[0];
            b.y = bp[D];
            acc[t] = __builtin_amdgcn_wmma_f32_16x16x4_f32(
                false, a, false, b, (short)0, acc[t], false, false);
        }
    }

#pragma unroll
    for (int t = 0; t < NT; ++t) {
        float* cp = C + (size_t)(m0 + 8 * half) * D + t * 16 + lm;
#pragma unroll
        for (int v = 0; v < 8; ++v) cp[(size_t)v * D] = acc[t][v];
    }
}

// ---------------------------------------------------------------------------
// Fused normalized scatter-add (SpMM): one wave per edge.
//   out[dst] += attn[which] * rin[dst] * rout[src] * H[src]
// Lane j gathers D/32 contiguous floats (128-bit load for D=128) — the wave
// covers a full feature row coalesced — then issues f32 global atomics.
// ---------------------------------------------------------------------------
template <int D>
__global__ __launch_bounds__(256)
void spmm_scatter(const float* __restrict__ H,
                  const int* __restrict__ src, const int* __restrict__ dst,
                  const float* __restrict__ rout, const float* __restrict__ rin,
                  const float* __restrict__ attn, int which,
                  float* __restrict__ out, int E) {
    int gid  = blockIdx.x * blockDim.x + threadIdx.x;
    int e    = gid >> 5;
    int lane = gid & 31;
    if (e >= E) return;

    int s = src[e];
    int d = dst[e];
    float c = attn[which] * rout[s] * rin[d];

    constexpr int V = D / 32;                 // 4 (D=128) or 2 (D=64) floats/lane
    const float* hp = H + (size_t)s * D + lane * V;
    float*       op = out + (size_t)d * D + lane * V;

    float hv[V];
#pragma unroll
    for (int j = 0; j < V; ++j) hv[j] = hp[j];   // merges into b128/b64 load
#pragma unroll
    for (int j = 0; j < V; ++j) atomicAdd(&op[j], c * hv[j]);
}

// ---------------------------------------------------------------------------
// Launch
// ---------------------------------------------------------------------------
extern "C" void kernel_launch(void* const* d_in, const int* in_sizes, int n_in,
                              void* d_out, int out_size, void* d_ws, size_t ws_size,
                              hipStream_t stream) {
    const float* X    = (const float*)d_in[0];
    const int*   src1 = (const int*)  d_in[1];
    const int*   dst1 = (const int*)  d_in[2];
    const int*   src2 = (const int*)  d_in[3];
    const int*   dst2 = (const int*)  d_in[4];
    const float* attn = (const float*)d_in[5];
    const float* W1   = (const float*)d_in[6];
    const float* b1   = (const float*)d_in[7];
    const float* W2   = (const float*)d_in[8];
    const float* b2   = (const float*)d_in[9];
    float* out = (float*)d_out;

    const int N = in_sizes[0] / D_IN;   // 50000 (multiple of 16)
    const int E = in_sizes[1];          // 600000

    // Workspace carve-out (floats): 4 rsqrt-degree arrays, H1, x1, H2
    float* ws    = (float*)d_ws;
    float* rout1 = ws;  ws += N;
    float* rin1  = ws;  ws += N;
    float* rout2 = ws;  ws += N;
    float* rin2  = ws;  ws += N;
    float* H1    = ws;  ws += (size_t)N * D_HID;
    float* x1    = ws;  ws += (size_t)N * D_HID;
    float* H2    = ws;  ws += (size_t)N * D_OUT;

    const int T = 256;
    // --- degrees ---
    zero_f32<<<(4 * N + T - 1) / T, T, 0, stream>>>(rout1, 4 * N);
    degree_kernel<<<(E + T - 1) / T, T, 0, stream>>>(src1, dst1, src2, dst2,
                                                     rout1, rin1, rout2, rin2, E);
    rsqrt_clamp<<<(4 * N + T - 1) / T, T, 0, stream>>>(rout1, 4 * N);

    // --- layer 1: H1 = X @ W1 ; x1 = (a0+a1)*b1 + fused scatter of both graphs ---
    gemm_wmma_f32<D_HID / 16><<<N / 16, 32, 0, stream>>>(X, W1, H1, D_IN, D_HID);
    {
        int total = N * D_HID;
        bias_fill<<<(total + T - 1) / T, T, 0, stream>>>(x1, b1, attn, total, D_HID - 1);
    }
    {
        long long threads = (long long)E * 32;
        int blocks = (int)((threads + T - 1) / T);
        spmm_scatter<D_HID><<<blocks, T, 0, stream>>>(H1, src1, dst1, rout1, rin1, attn, 0, x1, E);
        spmm_scatter<D_HID><<<blocks, T, 0, stream>>>(H1, src2, dst2, rout2, rin2, attn, 1, x1, E);
    }

    // --- layer 2: H2 = x1 @ W2 ; out = (a0+a1)*b2 + fused scatter of both graphs ---
    gemm_wmma_f32<D_OUT / 16><<<N / 16, 32, 0, stream>>>(x1, W2, H2, D_HID, D_OUT);
    {
        int total = N * D_OUT;
        bias_fill<<<(total + T - 1) / T, T, 0, stream>>>(out, b2, attn, total, D_OUT - 1);
    }
    {
        long long threads = (long long)E * 32;
        int blocks = (int)((threads + T - 1) / T);
        spmm_scatter<D_OUT><<<blocks, T, 0, stream>>>(H2, src1, dst1, rout1, rin1, attn, 0, out, E);
        spmm_scatter<D_OUT><<<blocks, T, 0, stream>>>(H2, src2, dst2, rout2, rin2, attn, 1, out, E);
    }
}